// STGCN_62045097558531
// MI455X (gfx1250) — compile-verified
//
#include <hip/hip_runtime.h>
#include <math.h>

typedef __attribute__((ext_vector_type(2))) float v2f;
typedef __attribute__((ext_vector_type(8))) float v8f;

#define B_   128
#define NS_  128
#define CIN_ 2
#define T0_  9
#define CT_  64
#define CS_  16
#define N_   (B_ * NS_)      // 16384
#define E_   (N_ * 16)       // 262144
#define D_   (NS_ * CT_)     // 8192

static __device__ __forceinline__ v8f wmma_f32(v2f a, v2f b, v8f c) {
  // V_WMMA_F32_16X16X4_F32 : D = A(16x4,f32) x B(4x16,f32) + C(16x16,f32)
  return __builtin_amdgcn_wmma_f32_16x16x4_f32(
      /*neg_a=*/false, a, /*neg_b=*/false, b,
      /*c_mod=*/(short)0, c, /*reuse_a=*/false, /*reuse_b=*/false);
}

// ---------------------------------------------------------------------------
// small helpers
// ---------------------------------------------------------------------------
__global__ void k_fill(float* __restrict__ p, float v, long n) {
  long gid = (long)blockIdx.x * blockDim.x + threadIdx.x;
  if (gid < n) p[gid] = v;
}

__global__ void k_deg_accum(const int* __restrict__ cols,
                            const float* __restrict__ ew,
                            float* __restrict__ deg) {
  int e = blockIdx.x * blockDim.x + threadIdx.x;
  if (e < E_) atomicAdd(&deg[cols[e]], ew[e]);
}

// per-edge normalized coefficient: dis[row]*ew*dis[col]
__global__ void k_coef(const int* __restrict__ rows,
                       const int* __restrict__ cols,
                       const float* __restrict__ ew,
                       const float* __restrict__ deg,
                       float* __restrict__ coef) {
  int e = blockIdx.x * blockDim.x + threadIdx.x;
  if (e < E_) coef[e] = rsqrtf(deg[rows[e]]) * ew[e] * rsqrtf(deg[cols[e]]);
}

// conv weight transpose: w[o, kappa] (o<128, kappa<Ktot)  ->  wT[kappa, o]
__global__ void k_wtrans(const float* __restrict__ w, float* __restrict__ wT,
                         int Ktot) {
  int gid = blockIdx.x * blockDim.x + threadIdx.x;
  if (gid >= Ktot * 128) return;
  int o = gid & 127;
  int k = gid >> 7;
  wT[gid] = w[(size_t)o * Ktot + k];
}

// bias + relu over a column-major [Nout, 128] buffer (in place)
__global__ void k_bias_relu_colmajor(float* __restrict__ cT,
                                     const float* __restrict__ bias, long n) {
  long gid = (long)blockIdx.x * blockDim.x + threadIdx.x;
  if (gid < n) {
    int col = (int)(gid >> 7);               // gid = col*128 + row
    cT[gid] = fmaxf(cT[gid] + bias[col], 0.f);
  }
}

// bias add over a row-major [rows, Nout] buffer (in place, no relu)
__global__ void k_bias_add_rowmajor(float* __restrict__ c,
                                    const float* __restrict__ bias,
                                    int Nout, long n) {
  long gid = (long)blockIdx.x * blockDim.x + threadIdx.x;
  if (gid < n) {
    int col = (int)(gid % Nout);
    c[gid] += bias[col];
  }
}

// ---------------------------------------------------------------------------
// Time1dConvGLU as WMMA GEMM (compile-time shapes -> branchless inner loop):
//   out[n, c, t0] = p * sigmoid(q)
//   p = sum_{kappa} in[n, kappa/3, t0 + kappa%3] * wT[kappa, c]      + b[c]
//   q =                       "                  * wT[kappa, c+CT]   + b[c+CT]
// One wave: one 16(node) x 16(channel) tile pair (p & q) at one t0.
// ---------------------------------------------------------------------------
template <int Cin, int Tin, int Tout>
__global__ void k_conv_glu_wmma(const float* __restrict__ in,
                                const float* __restrict__ wT,   // [Ktot,128]
                                const float* __restrict__ bias,
                                float* __restrict__ out) {
  constexpr int Ktot = Cin * 3;
  constexpr int Ksteps = (Ktot + 3) / 4;

  long tid = (long)blockIdx.x * blockDim.x + threadIdx.x;
  int wave = (int)(tid >> 5);
  constexpr int totalWaves = (N_ / 16) * (CT_ / 16) * Tout;
  if (wave >= totalWaves) return;

  int t0 = wave % Tout;
  int j  = (wave / Tout) % (CT_ / 16);
  int m0 = wave / (Tout * (CT_ / 16));

  int lane = (int)(tid & 31);
  int half = lane >> 4;
  int r    = lane & 15;

  int m  = m0 * 16 + r;            // node row held by this lane (A matrix)
  int op = j * 16 + r;             // p-output channel (B/C column)

  const float* arow = in + (size_t)m * Cin * Tin + t0;

  v8f accp = {0.f, 0.f, 0.f, 0.f, 0.f, 0.f, 0.f, 0.f};
  v8f accq = {0.f, 0.f, 0.f, 0.f, 0.f, 0.f, 0.f, 0.f};

#pragma unroll
  for (int s = 0; s < Ksteps; ++s) {
    const int kl = 4 * s;          // kappa for lanes 0-15 (.x)
    const int kh = 4 * s + 2;      // kappa for lanes 16-31 (.x)
    int ka = half ? kh : kl;       // runtime select of two constants
    // compile-time (i,k) offsets for both halves; one cndmask at runtime
    int offa0 = half ? ((kh / 3) * Tin + kh % 3) : ((kl / 3) * Tin + kl % 3);
    int offa1 = half ? (((kh + 1) / 3) * Tin + (kh + 1) % 3)
                     : (((kl + 1) / 3) * Tin + (kl + 1) % 3);
    bool v0 = ka < Ktot;           // folds to true when Ktot % 4 == 0
    bool v1 = (ka + 1) < Ktot;

    v2f a, bp, bq;
    float t;
    t = arow[v0 ? offa0 : 0];                     a.x = v0 ? t : 0.f;
    t = arow[v1 ? offa1 : 0];                     a.y = v1 ? t : 0.f;
    const float* w0 = wT + (size_t)(v0 ? ka : 0) * 128;
    const float* w1 = wT + (size_t)(v1 ? (ka + 1) : 0) * 128;
    t = w0[op];                                   bp.x = v0 ? t : 0.f;
    t = w1[op];                                   bp.y = v1 ? t : 0.f;
    t = w0[op + CT_];                             bq.x = v0 ? t : 0.f;
    t = w1[op + CT_];                             bq.y = v1 ? t : 0.f;

    accp = wmma_f32(a, bp, accp);
    accq = wmma_f32(a, bq, accq);
  }

  float bvp = bias[op];
  float bvq = bias[op + CT_];
#pragma unroll
  for (int r8 = 0; r8 < 8; ++r8) {
    int row = m0 * 16 + r8 + 8 * half;            // node index of acc element
    float p = accp[r8] + bvp;
    float q = accq[r8] + bvq;
    float sg = 1.f / (1.f + __expf(-q));
    out[(size_t)row * CT_ * Tout + (size_t)op * Tout + t0] = p * sg;
  }
}

// ---------------------------------------------------------------------------
// GCN channel projection xw[n,o,t] = sum_i in[n,i,t] * wg[i,o]  (K=64, Nout=16)
// ---------------------------------------------------------------------------
__global__ void k_gcn_xw_wmma(const float* __restrict__ in,
                              const float* __restrict__ wg,
                              float* __restrict__ xw, int T) {
  long tid = (long)blockIdx.x * blockDim.x + threadIdx.x;
  int wave = (int)(tid >> 5);
  int totalWaves = (N_ / 16) * T;
  if (wave >= totalWaves) return;

  int t  = wave % T;
  int m0 = wave / T;
  int lane = (int)(tid & 31);
  int half = lane >> 4;
  int r    = lane & 15;
  int m    = m0 * 16 + r;

  const float* arow = in + (size_t)m * CT_ * T + t;

  v8f acc = {0.f, 0.f, 0.f, 0.f, 0.f, 0.f, 0.f, 0.f};
#pragma unroll 4
  for (int s = 0; s < CT_ / 4; ++s) {
    int k0 = s * 4 + 2 * half;
    v2f a, b;
    a.x = arow[(size_t)k0 * T];
    a.y = arow[(size_t)(k0 + 1) * T];
    b.x = wg[(size_t)k0 * CS_ + r];
    b.y = wg[(size_t)(k0 + 1) * CS_ + r];
    acc = wmma_f32(a, b, acc);
  }
#pragma unroll
  for (int r8 = 0; r8 < 8; ++r8) {
    int row = m0 * 16 + r8 + 8 * half;
    xw[(size_t)row * CS_ * T + (size_t)r * T + t] = acc[r8];
  }
}

// ---------------------------------------------------------------------------
// Edge scatter: agg[col] += xw[row] * coef[e]
// ---------------------------------------------------------------------------
__global__ void k_gcn_scatter(const float* __restrict__ xw,
                              const int* __restrict__ rows,
                              const int* __restrict__ cols,
                              const float* __restrict__ coef,
                              float* __restrict__ agg, int T) {
  long gid = (long)blockIdx.x * blockDim.x + threadIdx.x;
  int per = CS_ * T;
  long total = (long)E_ * per;
  if (gid >= total) return;
  int e = (int)(gid / per);
  int j = (int)(gid - (long)e * per);
  int rn = rows[e], cn = cols[e];
  atomicAdd(&agg[(size_t)cn * per + j], xw[(size_t)rn * per + j] * coef[e]);
}

__global__ void k_gcn_finish(const float* __restrict__ xw,
                             const float* __restrict__ agg,
                             const float* __restrict__ deg,
                             const float* __restrict__ bias,
                             float* __restrict__ out, int T) {
  long gid = (long)blockIdx.x * blockDim.x + threadIdx.x;
  long total = (long)N_ * CS_ * T;
  if (gid >= total) return;
  int per = CS_ * T;
  int n = (int)(gid / per);
  int j = (int)(gid - (long)n * per);
  int o = j / T;
  float v = agg[gid] + xw[gid] / deg[n] + bias[o];
  out[gid] = fmaxf(v, 0.f);
}

// ---------------------------------------------------------------------------
// LayerNorm over (NS, T) per (graph b, channel c); affine g/beta are [NS, T].
// OUT_T: write K-major H for the MLP: H[(s*CT + c) * 128 + b]   (T must be 1)
// ---------------------------------------------------------------------------
template <bool OUT_T>
__global__ void k_layernorm(const float* __restrict__ in,
                            const float* __restrict__ g,
                            const float* __restrict__ be,
                            float* __restrict__ out,
                            int C, int T) {
  int b = blockIdx.x;
  int c = blockIdx.y;
  int tid = threadIdx.x;
  int M = NS_ * T;

  __shared__ float red0[128];
  __shared__ float red1[128];

  float sum = 0.f, sq = 0.f;
  for (int idx = tid; idx < M; idx += blockDim.x) {
    int s = idx / T, t = idx - s * T;
    float v = in[((size_t)(b * NS_ + s) * C + c) * T + t];
    sum += v; sq += v * v;
  }
  red0[tid] = sum; red1[tid] = sq;
  __syncthreads();
  for (int off = 64; off > 0; off >>= 1) {
    if (tid < off) { red0[tid] += red0[tid + off]; red1[tid] += red1[tid + off]; }
    __syncthreads();
  }
  float mu = red0[0] / (float)M;
  float var = red1[0] / (float)M - mu * mu;
  float rstd = rsqrtf(var + 1e-5f);

  for (int idx = tid; idx < M; idx += blockDim.x) {
    int s = idx / T, t = idx - s * T;
    float v = in[((size_t)(b * NS_ + s) * C + c) * T + t];
    float y = (v - mu) * rstd * g[idx] + be[idx];
    if (OUT_T)
      out[((size_t)s * CT_ + c) * B_ + b] = y;   // K-major H, T == 1
    else
      out[((size_t)(b * NS_ + s) * C + c) * T + t] = y;
  }
}

// ---------------------------------------------------------------------------
// K-split WMMA GEMM: out += A[128, K-chunk] @ Bw[K-chunk, Nout]
// A given K-major AT[K,128]. Wave = (split, 16-wide N-strip); each wave owns
// all 8 M-tiles so Bw streams from HBM exactly once in total, while the
// K-split raises wave count ~16x to cover the HBM bandwidth-delay product
// (23.3 TB/s x ~1us needs ~20 MB in flight; 512 waves can't hold that).
// Partial tiles are combined with global_atomic_add_f32; bias/activation is
// applied by a separate tiny epilogue kernel.
// STORE_T: accumulate into K-major [Nout,128]; else row-major [128,Nout].
// ---------------------------------------------------------------------------
template <bool STORE_T>
__global__ void k_gemm_wmma_split(const float* __restrict__ AT,   // [K,128]
                                  const float* __restrict__ Bw,   // [K,Nout]
                                  float* __restrict__ out,
                                  int K, int Nout, int kchunk) {
  long tid = (long)blockIdx.x * blockDim.x + threadIdx.x;
  int wave = (int)(tid >> 5);
  int nstrips = Nout >> 4;
  int nsplit = K / kchunk;
  if (wave >= nstrips * nsplit) return;

  int strip = wave % nstrips;
  int split = wave / nstrips;
  int kbeg = split * kchunk;
  int kend = kbeg + kchunk;

  int lane = (int)(tid & 31);
  int half = lane >> 4;
  int r    = lane & 15;
  int ncol = strip * 16 + r;

  v8f acc[8];
#pragma unroll
  for (int i = 0; i < 8; ++i) {
    v8f z = {0.f, 0.f, 0.f, 0.f, 0.f, 0.f, 0.f, 0.f};
    acc[i] = z;
  }

#pragma unroll 2
  for (int k0 = kbeg; k0 < kend; k0 += 4) {
    int k = k0 + 2 * half;
    const float* arow0 = AT + (size_t)k * B_;
    const float* arow1 = arow0 + B_;
    v2f b;
    b.x = Bw[(size_t)k * Nout + ncol];
    b.y = Bw[(size_t)(k + 1) * Nout + ncol];
#pragma unroll
    for (int mt = 0; mt < 8; ++mt) {
      v2f a;
      a.x = arow0[mt * 16 + r];
      a.y = arow1[mt * 16 + r];
      acc[mt] = wmma_f32(a, b, acc[mt]);
    }
  }

#pragma unroll
  for (int mt = 0; mt < 8; ++mt) {
#pragma unroll
    for (int r8 = 0; r8 < 8; ++r8) {
      int row = mt * 16 + r8 + 8 * half;
      if (STORE_T)
        atomicAdd(&out[(size_t)ncol * B_ + row], acc[mt][r8]);
      else
        atomicAdd(&out[(size_t)row * Nout + ncol], acc[mt][r8]);
    }
  }
}

// ---------------------------------------------------------------------------
// Host-side orchestration
// ---------------------------------------------------------------------------
extern "C" void kernel_launch(void* const* d_in, const int* in_sizes, int n_in,
                              void* d_out, int out_size, void* d_ws, size_t ws_size,
                              hipStream_t stream) {
  (void)in_sizes; (void)n_in; (void)out_size; (void)ws_size;

  const float* x      = (const float*)d_in[0];
  const int*   ei     = (const int*)d_in[1];
  const float* ew     = (const float*)d_in[2];
  const float* w_t1a  = (const float*)d_in[3];
  const float* b_t1a  = (const float*)d_in[4];
  const float* w_g1   = (const float*)d_in[5];
  const float* b_g1   = (const float*)d_in[6];
  const float* w_t1b  = (const float*)d_in[7];
  const float* b_t1b  = (const float*)d_in[8];
  const float* g_ln1  = (const float*)d_in[9];
  const float* be_ln1 = (const float*)d_in[10];
  const float* w_t2a  = (const float*)d_in[11];
  const float* b_t2a  = (const float*)d_in[12];
  const float* w_g2   = (const float*)d_in[13];
  const float* b_g2   = (const float*)d_in[14];
  const float* w_t2b  = (const float*)d_in[15];
  const float* b_t2b  = (const float*)d_in[16];
  const float* g_ln2  = (const float*)d_in[17];
  const float* be_ln2 = (const float*)d_in[18];
  const float* w_o1   = (const float*)d_in[19];
  const float* b_o1   = (const float*)d_in[20];
  const float* w_o2   = (const float*)d_in[21];
  const float* b_o2   = (const float*)d_in[22];
  float* out = (float*)d_out;
  float* ws  = (float*)d_ws;

  // workspace layout (floats); region A is recycled through the pipeline
  float* hA    = ws + 0;            // [N,64,7] = 7,340,032
  float* xw    = ws + 7340032;      // [N,16,7] = 1,835,008 (reused for T=3)
  float* agg   = ws + 9175040;      // [N,16,7]            (reused for T=3)
  float* gout  = ws + 11010048;     // [N,16,7]            (reused for T=3)
  float* hB    = ws + 12845056;     // [N,64,5] = 5,242,880 (reused for hC [N,64,3])
  float* deg   = ws + 18087936;     // [N]
  float* coef  = ws + 18104320;     // [E]
  float* wt1a  = ws + 18366464;     // [6,128]
  float* wt1b  = ws + 18367232;     // [48,128]
  float* wt2a  = ws + 18373376;     // [192,128]
  float* wt2b  = ws + 18397952;     // [48,128]
  float* ln1   = hA;                // [N,64,5]   (hA dead)
  float* hC    = hB;                // [N,64,3]   (hB dead)
  float* hD    = hA;                // [N,64,1]   (ln1 dead)
  float* H_T   = ws + 1048576;      // [8192,128] K-major H (ln1 tail dead)
  float* C1_T  = ws + 2097152;      // [8192,128] K-major relu(H@w_o1+b)

  const int* rows = ei;
  const int* cols = ei + E_;

  auto cdiv = [](long a, long b) { return (unsigned)((a + b - 1) / b); };

  // ---- one-time per call: weight transposes, degrees, edge coefficients
  k_wtrans<<<cdiv(6 * 128, 256), 256, 0, stream>>>(w_t1a, wt1a, 6);
  k_wtrans<<<cdiv(48 * 128, 256), 256, 0, stream>>>(w_t1b, wt1b, 48);
  k_wtrans<<<cdiv(192 * 128, 256), 256, 0, stream>>>(w_t2a, wt2a, 192);
  k_wtrans<<<cdiv(48 * 128, 256), 256, 0, stream>>>(w_t2b, wt2b, 48);
  k_fill<<<cdiv(N_, 256), 256, 0, stream>>>(deg, 1.0f, N_);
  k_deg_accum<<<cdiv(E_, 256), 256, 0, stream>>>(cols, ew, deg);
  k_coef<<<cdiv(E_, 256), 256, 0, stream>>>(rows, cols, ew, deg, coef);

  // ---- block 1
  long wv = (long)(N_ / 16) * (CT_ / 16) * 7;
  k_conv_glu_wmma<CIN_, 9, 7><<<cdiv(wv * 32, 256), 256, 0, stream>>>(x, wt1a, b_t1a, hA);

  wv = (long)(N_ / 16) * 7;
  k_gcn_xw_wmma<<<cdiv(wv * 32, 256), 256, 0, stream>>>(hA, w_g1, xw, 7);
  k_fill<<<cdiv((long)N_ * CS_ * 7, 256), 256, 0, stream>>>(agg, 0.f, (long)N_ * CS_ * 7);
  k_gcn_scatter<<<cdiv((long)E_ * CS_ * 7, 256), 256, 0, stream>>>(xw, rows, cols, coef, agg, 7);
  k_gcn_finish<<<cdiv((long)N_ * CS_ * 7, 256), 256, 0, stream>>>(xw, agg, deg, b_g1, gout, 7);

  wv = (long)(N_ / 16) * (CT_ / 16) * 5;
  k_conv_glu_wmma<CS_, 7, 5><<<cdiv(wv * 32, 256), 256, 0, stream>>>(gout, wt1b, b_t1b, hB);

  k_layernorm<false><<<dim3(B_, CT_), 128, 0, stream>>>(hB, g_ln1, be_ln1, ln1, CT_, 5);

  // ---- block 2
  wv = (long)(N_ / 16) * (CT_ / 16) * 3;
  k_conv_glu_wmma<CT_, 5, 3><<<cdiv(wv * 32, 256), 256, 0, stream>>>(ln1, wt2a, b_t2a, hC);

  wv = (long)(N_ / 16) * 3;
  k_gcn_xw_wmma<<<cdiv(wv * 32, 256), 256, 0, stream>>>(hC, w_g2, xw, 3);
  k_fill<<<cdiv((long)N_ * CS_ * 3, 256), 256, 0, stream>>>(agg, 0.f, (long)N_ * CS_ * 3);
  k_gcn_scatter<<<cdiv((long)E_ * CS_ * 3, 256), 256, 0, stream>>>(xw, rows, cols, coef, agg, 3);
  k_gcn_finish<<<cdiv((long)N_ * CS_ * 3, 256), 256, 0, stream>>>(xw, agg, deg, b_g2, gout, 3);

  wv = (long)(N_ / 16) * (CT_ / 16) * 1;
  k_conv_glu_wmma<CS_, 3, 1><<<cdiv(wv * 32, 256), 256, 0, stream>>>(gout, wt2b, b_t2b, hD);

  k_layernorm<true><<<dim3(B_, CT_), 128, 0, stream>>>(hD, g_ln2, be_ln2, H_T, CT_, 1);

  // ---- output MLP (bandwidth-bound on w_o1: 256 MB streamed exactly once;
  //      K-split x16 / x64 for enough in-flight loads to cover the BDP)
  k_fill<<<cdiv((long)D_ * B_, 256), 256, 0, stream>>>(C1_T, 0.f, (long)D_ * B_);
  {
    long waves = (long)(D_ / 16) * 16;                       // 512 strips x 16 splits
    k_gemm_wmma_split<true><<<cdiv(waves * 32, 256), 256, 0, stream>>>(
        H_T, w_o1, C1_T, D_, D_, D_ / 16);                   // kchunk = 512
  }
  k_bias_relu_colmajor<<<cdiv((long)D_ * B_, 256), 256, 0, stream>>>(
      C1_T, b_o1, (long)D_ * B_);

  k_fill<<<cdiv((long)B_ * NS_ * CIN_, 256), 256, 0, stream>>>(out, 0.f, (long)B_ * NS_ * CIN_);
  {
    const int Nout2 = NS_ * CIN_;                            // 256
    long waves = (long)(Nout2 / 16) * 64;                    // 16 strips x 64 splits
    k_gemm_wmma_split<false><<<cdiv(waves * 32, 256), 256, 0, stream>>>(
        C1_T, w_o2, out, D_, Nout2, D_ / 64);                // kchunk = 128
  }
  k_bias_add_rowmajor<<<cdiv((long)B_ * NS_ * CIN_, 256), 256, 0, stream>>>(
      out, b_o2, NS_ * CIN_, (long)B_ * NS_ * CIN_);
}